// Decoder_4561255269164
// MI455X (gfx1250) — compile-verified
//
#include <hip/hip_runtime.h>
#include <hip/hip_bf16.h>
#include <math.h>

typedef float v2f __attribute__((ext_vector_type(2)));
typedef float v8f __attribute__((ext_vector_type(8)));

#define BIGD 1e30f

__device__ __forceinline__ unsigned f2key(float f) {
    unsigned u = __float_as_uint(f);
    return (u & 0x80000000u) ? ~u : (u | 0x80000000u);
}
__device__ __forceinline__ float key2f(unsigned k) {
    unsigned u = (k & 0x80000000u) ? (k & 0x7FFFFFFFu) : ~k;
    return __uint_as_float(u);
}

// ---------------------------------------------------------------------------
// Kernel 1: zero accumulators, squared norms, local-max mask, sortable keys,
// init packed (d2,idx) running minima.
// ---------------------------------------------------------------------------
__global__ void init_kernel(const float* __restrict__ pred,
                            const float* __restrict__ cand_xyz,
                            const float* __restrict__ tgt_xyz,
                            unsigned* __restrict__ keys, int* __restrict__ lmax,
                            float* __restrict__ cand2, float* __restrict__ tgt2,
                            float* __restrict__ num, float* __restrict__ den,
                            int* __restrict__ exactFlag, float* __restrict__ exactRGB,
                            unsigned long long* __restrict__ bestB,   // per target
                            unsigned long long* __restrict__ bestF,   // per candidate
                            float* __restrict__ d_out, int L, int N) {
    const unsigned long long INIT =
        ((unsigned long long)__float_as_uint(BIGD) << 32) | 0xFFFFFFFFull;
    int tid = blockIdx.x * blockDim.x + threadIdx.x;
    int stride = gridDim.x * blockDim.x;
    for (int i = tid; i < L; i += stride) {
        float x = cand_xyz[3 * i], y = cand_xyz[3 * i + 1], z = cand_xyz[3 * i + 2];
        cand2[i] = x * x + y * y + z * z;
        den[i] = 0.0f;
        exactFlag[i] = 0;
        num[3 * i] = 0.0f; num[3 * i + 1] = 0.0f; num[3 * i + 2] = 0.0f;
        exactRGB[3 * i] = 0.0f; exactRGB[3 * i + 1] = 0.0f; exactRGB[3 * i + 2] = 0.0f;
        bestF[i] = INIT;
    }
    for (int i = tid; i < N; i += stride) {
        float x = tgt_xyz[3 * i], y = tgt_xyz[3 * i + 1], z = tgt_xyz[3 * i + 2];
        tgt2[i] = x * x + y * y + z * z;
        bestB[i] = INIT;
    }
    for (int g = tid; g < (L >> 3); g += stride) {
        int base = g * 8;
        int bi = 0; float bv = pred[base];
        #pragma unroll
        for (int j = 1; j < 8; ++j) {
            float v = pred[base + j];
            if (v > bv) { bv = v; bi = j; }
        }
        #pragma unroll
        for (int j = 0; j < 8; ++j) {
            int l = base + j;
            int im = (j == bi) ? 1 : 0;
            lmax[l] = im;
            keys[l] = f2key(im ? __builtin_inff() : pred[l]);
        }
    }
    if (tid == 0) { d_out[0] = 0.0f; d_out[1] = 0.0f; }
}

// ---------------------------------------------------------------------------
// Kernel 2 (single block): 4-pass radix select for k-th smallest, keep mask,
// BCE coord loss.
// ---------------------------------------------------------------------------
__global__ void select_kernel(const float* __restrict__ pred,
                              const int* __restrict__ keep_target,
                              const int* __restrict__ points_num,
                              const unsigned* __restrict__ keys,
                              const int* __restrict__ lmax,
                              int* __restrict__ keep,
                              float* __restrict__ d_out, int L) {
    __shared__ unsigned hist[256];
    __shared__ unsigned sPrefix;
    __shared__ int sR;
    __shared__ float sThr;
    __shared__ float red[256];
    const int tid = threadIdx.x;
    const int nthr = blockDim.x;

    if (tid == 0) { sPrefix = 0u; sR = L - points_num[0] - 1; }
    __syncthreads();

    for (int pass = 0; pass < 4; ++pass) {
        const int shift = 24 - 8 * pass;
        const unsigned himask = (pass == 0) ? 0u : (0xFFFFFFFFu << (shift + 8));
        for (int b = tid; b < 256; b += nthr) hist[b] = 0u;
        __syncthreads();
        const unsigned pfx = sPrefix;
        for (int l = tid; l < L; l += nthr) {
            unsigned k = keys[l];
            if ((k & himask) == pfx) atomicAdd(&hist[(k >> shift) & 0xFFu], 1u);
        }
        __syncthreads();
        if (tid == 0) {
            unsigned cum = 0u;
            int r = sR;
            for (int b = 0; b < 256; ++b) {
                unsigned h = hist[b];
                if (cum + h > (unsigned)r) {
                    sPrefix = pfx | ((unsigned)b << shift);
                    sR = r - (int)cum;
                    break;
                }
                cum += h;
            }
        }
        __syncthreads();
    }
    if (tid == 0) sThr = key2f(sPrefix);
    __syncthreads();
    const float thr = sThr;

    float acc = 0.0f;
    for (int l = tid; l < L; l += nthr) {
        float p = pred[l];
        keep[l] = ((p > thr) || lmax[l]) ? 1 : 0;
        float t = (float)keep_target[l];
        acc += fmaxf(p, 0.0f) - p * t + log1pf(expf(-fabsf(p)));
    }
    red[tid] = acc;
    __syncthreads();
    for (int s = nthr >> 1; s > 0; s >>= 1) {
        if (tid < s) red[tid] += red[tid + s];
        __syncthreads();
    }
    if (tid == 0) d_out[0] = red[0];
}

// ---------------------------------------------------------------------------
// Kernel 3: build packed B operands: per (tile, lane) a float4
//   {-2*coord[k0], -2*coord[k0+1] (K padded 3->4), norm2, maskAdd}
// maskAdd = 0 for usable columns, BIGD for masked/out-of-range. Three extra
// padding tiles at the end enable double-buffered unroll-2 loads in the loop.
// ---------------------------------------------------------------------------
__global__ void pack_kernel(const float* __restrict__ xyz,
                            const float* __restrict__ norm2,
                            const int* __restrict__ keepOrNull,
                            float4* __restrict__ packB,
                            int nCols, int nTilesPadded) {
    int total = nTilesPadded * 32;
    for (int i = blockIdx.x * blockDim.x + threadIdx.x; i < total;
         i += gridDim.x * blockDim.x) {
        int lane = i & 31;
        int t = i >> 5;
        int lm = lane & 15;
        int k0 = (lane >> 4) * 2;
        int col = t * 16 + lm;
        bool cv = col < nCols;
        int cc = cv ? col : 0;
        float bx = cv ? -2.0f * xyz[cc * 3 + k0] : 0.0f;
        float by = (cv && (k0 + 1 < 3)) ? -2.0f * xyz[cc * 3 + k0 + 1] : 0.0f;
        float cn = cv ? norm2[cc] : 0.0f;
        bool kp = cv && (keepOrNull ? (keepOrNull[cc] != 0) : true);
        packB[i] = make_float4(bx, by, cn, kp ? 0.0f : BIGD);
    }
}

// ---------------------------------------------------------------------------
// Kernel 4: generic tiled 1-NN via WMMA f32 16x16x4.
// One wave per (16-row tile, column chunk); unroll-2 double-buffered b128
// loads keep one prefetch in flight across each WMMA; branchless running
// argmin; shfl_xor butterfly; packed u64 atomicMin combine
// ((d2 bits << 32) | colIdx, low-index tie-break).
// ---------------------------------------------------------------------------
__global__ void nn_kernel(const float* __restrict__ rowXyz,
                          const float* __restrict__ rowNorm2,
                          const float4* __restrict__ packB,
                          unsigned long long* __restrict__ bestPack,
                          int nRows, int nTilesB) {
    const int lane = threadIdx.x & 31;
    const int wave = threadIdx.x >> 5;
    const int rt = blockIdx.x * (blockDim.x >> 5) + wave;
    const int nTilesM = (nRows + 15) >> 4;
    if (rt >= nTilesM) return;

    const int half = lane >> 4;
    const int lm = lane & 15;
    const int k0 = half * 2;

    // chunk of column tiles owned by this wave
    const int nChunks = gridDim.y;
    const int tilesPer = (nTilesB + nChunks - 1) / nChunks;
    const int t0 = blockIdx.y * tilesPer;
    const int t1 = min(t0 + tilesPer, nTilesB);

    // A operand: 16 rows of this tile (pre-loop, cold path)
    const int arow = rt * 16 + lm;
    const bool av = arow < nRows;
    const int ar = av ? arow : 0;
    v2f a;
    a.x = av ? rowXyz[ar * 3 + k0] : 0.0f;
    a.y = (av && (k0 + 1 < 3)) ? rowXyz[ar * 3 + k0 + 1] : 0.0f;

    float r2[8]; int mrow[8];
    #pragma unroll
    for (int v = 0; v < 8; ++v) {
        mrow[v] = rt * 16 + v + 8 * half;       // D row owned by (half, vgpr v)
        r2[v] = (mrow[v] < nRows) ? rowNorm2[mrow[v]] : 0.0f;
    }

    float bestD[8]; int bestI[8];
    #pragma unroll
    for (int v = 0; v < 8; ++v) { bestD[v] = BIGD; bestI[v] = 0x7FFFFFFF; }

    auto process = [&](const float4 pk, int t) {
        v2f b; b.x = pk.x; b.y = pk.y;
        v8f c = {0.f, 0.f, 0.f, 0.f, 0.f, 0.f, 0.f, 0.f};
        // D[m][n] = -2 * dot(row[m], col[n])   (m = v + 8*half, n = lm)
        v8f d = __builtin_amdgcn_wmma_f32_16x16x4_f32(false, a, false, b,
                                                      (short)0, c, false, false);
        const int col = (t << 4) | lm;
        const float cn = pk.z, madd = pk.w;
        #pragma unroll
        for (int v = 0; v < 8; ++v) {
            float d2 = fmaxf(r2[v] + cn + d[v], 0.0f) + madd;
            bool take = d2 < bestD[v];      // ascending col => strict < keeps lowest idx
            bestD[v] = take ? d2 : bestD[v];
            bestI[v] = take ? col : bestI[v];
        }
    };

    if (t0 < t1) {
        int t = t0;
        float4 pk0 = packB[(size_t)t * 32 + lane];
        float4 pk1 = packB[(size_t)(t + 1) * 32 + lane];        // padded: always safe
        for (; t + 1 < t1; t += 2) {
            const float4 pn0 = packB[(size_t)(t + 2) * 32 + lane];  // prefetch
            const float4 pn1 = packB[(size_t)(t + 3) * 32 + lane];  // prefetch
            process(pk0, t);
            process(pk1, t + 1);
            pk0 = pn0;
            pk1 = pn1;
        }
        if (t < t1) process(pk0, t);
    }

    // combine across the 16 lanes of each half (columns of the D tile)
    #pragma unroll
    for (int m = 1; m <= 8; m <<= 1) {
        #pragma unroll
        for (int v = 0; v < 8; ++v) {
            float od = __shfl_xor(bestD[v], m, 32);
            int oi = __shfl_xor(bestI[v], m, 32);
            bool take = (od < bestD[v]) || ((od == bestD[v]) && (oi < bestI[v]));
            bestD[v] = take ? od : bestD[v];
            bestI[v] = take ? oi : bestI[v];
        }
    }
    if (lm == 0) {
        #pragma unroll
        for (int v = 0; v < 8; ++v) {
            if (mrow[v] < nRows) {
                unsigned long long pk64 =
                    ((unsigned long long)__float_as_uint(bestD[v]) << 32) |
                    (unsigned)bestI[v];
                atomicMin(&bestPack[mrow[v]], pk64);
            }
        }
    }
}

// ---------------------------------------------------------------------------
// Kernel 5: consume backward 1-NN: weighted scatter / exact-match overwrite.
// ---------------------------------------------------------------------------
__global__ void scatter_backward_kernel(const unsigned long long* __restrict__ bestB,
                                        const float* __restrict__ tgt_rgb,
                                        float* __restrict__ num, float* __restrict__ den,
                                        int* __restrict__ exactFlag,
                                        float* __restrict__ exactRGB, int N) {
    for (int m = blockIdx.x * blockDim.x + threadIdx.x; m < N;
         m += gridDim.x * blockDim.x) {
        unsigned long long p = bestB[m];
        float d2 = __uint_as_float((unsigned)(p >> 32));
        int l = (int)(unsigned)(p & 0xFFFFFFFFull);
        if (d2 >= 1e29f) continue;           // no kept candidate found
        if (d2 <= 0.0f) {
            exactFlag[l] = 1;                // overwrite semantics, no weighted add
            exactRGB[3 * l + 0] = tgt_rgb[3 * m + 0];
            exactRGB[3 * l + 1] = tgt_rgb[3 * m + 1];
            exactRGB[3 * l + 2] = tgt_rgb[3 * m + 2];
        } else {
            float w = 1.0f / sqrtf(fmaxf(d2, 1e-30f));
            atomicAdd(&den[l], w);
            atomicAdd(&num[3 * l + 0], tgt_rgb[3 * m + 0] * w);
            atomicAdd(&num[3 * l + 1], tgt_rgb[3 * m + 1] * w);
            atomicAdd(&num[3 * l + 2], tgt_rgb[3 * m + 2] * w);
        }
    }
}

// ---------------------------------------------------------------------------
// Kernel 6: final masked L1 reduction into d_out[1]; forward color read
// straight out of the packed forward best.
// ---------------------------------------------------------------------------
__global__ void final_kernel(const float* __restrict__ cand_rgb,
                             const int* __restrict__ keep,
                             const float* __restrict__ num, const float* __restrict__ den,
                             const int* __restrict__ exactFlag,
                             const float* __restrict__ exactRGB,
                             const unsigned long long* __restrict__ bestF,
                             const float* __restrict__ tgt_rgb,
                             float* __restrict__ d_out, int L, int N) {
    __shared__ float red[256];
    float acc = 0.0f;
    for (int l = blockIdx.x * blockDim.x + threadIdx.x; l < L;
         l += gridDim.x * blockDim.x) {
        if (!keep[l]) continue;
        float rc0, rc1, rc2;
        if (exactFlag[l]) {
            rc0 = exactRGB[3 * l + 0]; rc1 = exactRGB[3 * l + 1]; rc2 = exactRGB[3 * l + 2];
        } else {
            const float dn = den[l];
            if (dn != 0.0f) {
                rc0 = num[3 * l + 0] / dn; rc1 = num[3 * l + 1] / dn; rc2 = num[3 * l + 2] / dn;
            } else {
                unsigned n = (unsigned)(bestF[l] & 0xFFFFFFFFull);
                n = (n < (unsigned)N) ? n : 0u;
                rc0 = tgt_rgb[3 * n + 0]; rc1 = tgt_rgb[3 * n + 1]; rc2 = tgt_rgb[3 * n + 2];
            }
        }
        acc += fabsf(cand_rgb[3 * l + 0] * 255.0f - rc0)
             + fabsf(cand_rgb[3 * l + 1] * 255.0f - rc1)
             + fabsf(cand_rgb[3 * l + 2] * 255.0f - rc2);
    }
    red[threadIdx.x] = acc;
    __syncthreads();
    for (int s = blockDim.x >> 1; s > 0; s >>= 1) {
        if (threadIdx.x < s) red[threadIdx.x] += red[threadIdx.x + s];
        __syncthreads();
    }
    if (threadIdx.x == 0) atomicAdd(&d_out[1], red[0]);
}

// ---------------------------------------------------------------------------
extern "C" void kernel_launch(void* const* d_in, const int* in_sizes, int n_in,
                              void* d_out_, int out_size, void* d_ws, size_t ws_size,
                              hipStream_t stream) {
    const float* pred        = (const float*)d_in[0];
    const float* cand_xyz    = (const float*)d_in[1];
    const float* cand_rgb    = (const float*)d_in[2];
    const float* tgt_xyz     = (const float*)d_in[3];
    const float* tgt_rgb     = (const float*)d_in[4];
    const int*   keep_target = (const int*)d_in[5];
    const int*   points_num  = (const int*)d_in[6];
    float* out = (float*)d_out_;

    const int L = in_sizes[0];
    const int N = in_sizes[3] / 3;
    const int nTilesL = (L + 15) / 16;
    const int nTilesN = (N + 15) / 16;
    const int PAD = 3;                       // extra column tiles for unroll-2 prefetch

    char* ws = (char*)d_ws;
    size_t off = 0;
    auto walloc = [&](size_t bytes) -> void* {
        void* p = (void*)(ws + off);
        off = (off + bytes + 255) & ~(size_t)255;
        return p;
    };
    unsigned*           keys   = (unsigned*)walloc((size_t)L * 4);
    int*                lmax   = (int*)walloc((size_t)L * 4);
    int*                keep   = (int*)walloc((size_t)L * 4);
    float*              cand2  = (float*)walloc((size_t)L * 4);
    float*              tgt2   = (float*)walloc((size_t)N * 4);
    float*              num    = (float*)walloc((size_t)L * 12);
    float*              den    = (float*)walloc((size_t)L * 4);
    int*                exactF = (int*)walloc((size_t)L * 4);
    float*              exactC = (float*)walloc((size_t)L * 12);
    unsigned long long* bestB  = (unsigned long long*)walloc((size_t)N * 8);
    unsigned long long* bestF  = (unsigned long long*)walloc((size_t)L * 8);
    float4*             packC  = (float4*)walloc((size_t)(nTilesL + PAD) * 32 * 16);
    float4*             packT  = (float4*)walloc((size_t)(nTilesN + PAD) * 32 * 16);

    init_kernel<<<64, 256, 0, stream>>>(pred, cand_xyz, tgt_xyz, keys, lmax, cand2, tgt2,
                                        num, den, exactF, exactC, bestB, bestF, out, L, N);
    select_kernel<<<1, 256, 0, stream>>>(pred, keep_target, points_num, keys, lmax, keep,
                                         out, L);
    pack_kernel<<<64, 256, 0, stream>>>(cand_xyz, cand2, keep, packC, L, nTilesL + PAD);
    pack_kernel<<<64, 256, 0, stream>>>(tgt_xyz, tgt2, nullptr, packT, N, nTilesN + PAD);

    const int wavesPerBlock = 8;            // 256 threads = 8 wave32
    const int nChunks = 8;                  // column-dimension split for occupancy

    {   // backward: rows = targets, cols = kept candidates
        dim3 grid((nTilesN + wavesPerBlock - 1) / wavesPerBlock, nChunks);
        nn_kernel<<<grid, 32 * wavesPerBlock, 0, stream>>>(tgt_xyz, tgt2, packC, bestB,
                                                           N, nTilesL);
    }
    scatter_backward_kernel<<<64, 256, 0, stream>>>(bestB, tgt_rgb, num, den, exactF,
                                                    exactC, N);
    {   // forward: rows = candidates, cols = targets
        dim3 grid((nTilesL + wavesPerBlock - 1) / wavesPerBlock, nChunks);
        nn_kernel<<<grid, 32 * wavesPerBlock, 0, stream>>>(cand_xyz, cand2, packT, bestF,
                                                           L, nTilesN);
    }
    final_kernel<<<64, 256, 0, stream>>>(cand_rgb, keep, num, den, exactF, exactC,
                                         bestF, tgt_rgb, out, L, N);
}